// LlamaLayer_59158879535501
// MI455X (gfx1250) — compile-verified
//
#include <hip/hip_runtime.h>
#include <math.h>

typedef float v2f __attribute__((ext_vector_type(2)));
typedef float v8f __attribute__((ext_vector_type(8)));
typedef int   v4i_t __attribute__((vector_size(16)));   // matches builtin pointee type

#define HIDDEN   4096
#define NHEAD    32
#define HDIM     128
#define BS       16
#define MB       128
#define KPAD     132      // LDS row stride (floats): (4*row + d) % 64 -> conflict-free WMMA-A reads
#define NW       2        // waves per block
#define NEG_BIG  (-1e30f)

#if __has_builtin(__builtin_amdgcn_global_load_async_to_lds_b128) && \
    __has_builtin(__builtin_amdgcn_s_wait_asynccnt)
#define USE_ASYNC 1
#else
#define USE_ASYNC 0
#endif

#define GAS1(p) ((__attribute__((address_space(1))) v4i_t*)(p))
#define LAS3(p) ((__attribute__((address_space(3))) v4i_t*)(p))

__device__ __forceinline__ void copy16_to_lds(const float* g, float* l) {
#if USE_ASYNC
    // ASYNCcnt-tracked direct global->LDS copy (16B per lane)
    __builtin_amdgcn_global_load_async_to_lds_b128(GAS1(g), LAS3(l), 0, 0);
#else
    *(float4*)l = *(const float4*)g;
#endif
}

#if USE_ASYNC
#define WAIT_ASYNC(n) do { __builtin_amdgcn_s_wait_asynccnt(n); asm volatile("" ::: "memory"); } while (0)
#else
#define WAIT_ASYNC(n) do { } while (0)
#endif

// ---------------- Kernel 1: RMSNorm -> q_norm (softmax scale folded in) ----------------
__global__ __launch_bounds__(256) void rmsnorm_q_kernel(
    const float* __restrict__ hidden, const float* __restrict__ w,
    float* __restrict__ qn)
{
    __shared__ float red[256];
    const int b = blockIdx.x;
    const int t = threadIdx.x;
    const float* x = hidden + (size_t)b * HIDDEN;

    float s = 0.f;
#pragma unroll
    for (int i = 0; i < HIDDEN / 256; ++i) {
        float v = x[t + i * 256];
        s += v * v;
    }
    red[t] = s;
    __syncthreads();
    for (int off = 128; off > 0; off >>= 1) {
        if (t < off) red[t] += red[t + off];
        __syncthreads();
    }
    const float inv = rsqrtf(red[0] / (float)HIDDEN + 1e-6f) * 0.08838834764831845f;
#pragma unroll
    for (int i = 0; i < HIDDEN / 256; ++i) {
        const int idx = t + i * 256;
        qn[(size_t)b * HIDDEN + idx] = x[idx] * inv * w[idx];
    }
}

// ---------------- Kernel 2: paged decode attention, WMMA QK^T, async K staging ----------------
__global__ __launch_bounds__(NW * 32) void paged_attn_kernel(
    const float* __restrict__ qn,
    const float* __restrict__ kcache,
    const float* __restrict__ vcache,
    const int*   __restrict__ btab,
    const int*   __restrict__ clen,
    const float* __restrict__ hidden,
    float*       __restrict__ out)
{
    __shared__ float sh_q[HDIM];
    __shared__ float sh_k[NW][2][BS * KPAD];   // double-buffered K tile per wave
    __shared__ float sh_acc[NW][HDIM];
    __shared__ float sh_m[NW];
    __shared__ float sh_l[NW];

    const int bh   = blockIdx.x;
    const int b    = bh >> 5;      // / NHEAD
    const int h    = bh & 31;      // % NHEAD
    const int t    = threadIdx.x;
    const int lane = t & 31;
    const int w    = t >> 5;

    sh_q[t]      = qn[(size_t)b * HIDDEN + h * HDIM + t];
    sh_q[t + 64] = qn[(size_t)b * HIDDEN + h * HDIM + t + 64];
    __syncthreads();

    const int ctx    = clen[b];
    const int ntiles = (ctx + BS - 1) / BS;                      // tile == KV block (BS=16)
    const int nit    = (ntiles > w) ? ((ntiles - w + NW - 1) / NW) : 0;

    const int row  = lane & 15;
    const int half = lane >> 4;

    float  m = NEG_BIG, l = 0.f;
    float4 acc = {0.f, 0.f, 0.f, 0.f};

    // Stage one K tile (16 rows x 128 floats, 512B coalesced per row) into sh_k[w][buf].
    auto stage = [&](int tile, int buf) {
        const int    blk   = btab[b * MB + tile];
        const size_t base0 = ((size_t)blk * BS * NHEAD + h) * HDIM;
        float* dst = &sh_k[w][buf][0];
#pragma unroll
        for (int j = 0; j < BS; ++j) {
            copy16_to_lds(&kcache[base0 + (size_t)j * (NHEAD * HDIM) + lane * 4],
                          &dst[j * KPAD + lane * 4]);
        }
    };

    if (nit > 0) stage(w, 0);

    for (int i = 0; i < nit; ++i) {
        const int  tile = w + i * NW;
        const int  p    = i & 1;
        const bool more = (i + 1 < nit);

        if (more) {
            stage(w + (i + 1) * NW, p ^ 1);   // prefetch next tile into other buffer
            WAIT_ASYNC(16);                   // in-order completion: older batch of 16 done
        } else {
            WAIT_ASYNC(0);
        }

        const float* ksh   = &sh_k[w][p][0];
        const int    blk   = btab[b * MB + tile];
        const size_t base0 = ((size_t)blk * BS * NHEAD + h) * HDIM;

        // scores(16x16, all columns identical) = K_tile(16x128) x q-replicated(128x16)
        v8f sc = {};
#pragma unroll
        for (int c = 0; c < HDIM / 4; ++c) {
            const int d0 = c * 4 + half * 2;
            v2f a, bq;
            a[0]  = ksh[row * KPAD + d0];
            a[1]  = ksh[row * KPAD + d0 + 1];
            bq[0] = sh_q[d0];
            bq[1] = sh_q[d0 + 1];
            sc = __builtin_amdgcn_wmma_f32_16x16x4_f32(
                false, a, false, bq, (short)0, sc, false, false);
        }

        // Broadcast the 16 row scores (rows 0-7 in lanes 0-15, rows 8-15 in lanes 16-31).
        const int sbase = tile * BS;
        float p16[16];
#pragma unroll
        for (int j = 0; j < 16; ++j) {
            const float v = __shfl(sc[j & 7], (j < 8) ? 0 : 16, 32);
            p16[j] = ((sbase + j) < ctx) ? v : NEG_BIG;
        }

        float tmax = p16[0];
#pragma unroll
        for (int j = 1; j < 16; ++j) tmax = fmaxf(tmax, p16[j]);

        const float mn   = fmaxf(m, tmax);
        const float corr = __expf(m - mn);

        float  ssum = 0.f;
        float4 pv   = {0.f, 0.f, 0.f, 0.f};
#pragma unroll
        for (int j = 0; j < 16; ++j) {
            const float e = __expf(p16[j] - mn);
            ssum += e;
            const float4 vv = *(const float4*)&vcache[base0 + (size_t)j * (NHEAD * HDIM) + lane * 4];
            pv.x += e * vv.x; pv.y += e * vv.y; pv.z += e * vv.z; pv.w += e * vv.w;
        }

        m = mn;
        l = l * corr + ssum;
        acc.x = acc.x * corr + pv.x;
        acc.y = acc.y * corr + pv.y;
        acc.z = acc.z * corr + pv.z;
        acc.w = acc.w * corr + pv.w;
    }

    // Cross-wave log-sum-exp merge.
    if (lane == 0) { sh_m[w] = m; sh_l[w] = l; }
    *(float4*)&sh_acc[w][lane * 4] = acc;
    __syncthreads();

    const float M = fmaxf(sh_m[0], sh_m[1]);
    float L = 0.f, a0 = 0.f, a1 = 0.f;
#pragma unroll
    for (int w2 = 0; w2 < NW; ++w2) {
        const float f = __expf(sh_m[w2] - M);
        L  += f * sh_l[w2];
        a0 += f * sh_acc[w2][t];
        a1 += f * sh_acc[w2][t + 64];
    }

    const size_t oidx = (size_t)b * HIDDEN + h * HDIM + t;
    out[oidx]      = hidden[oidx]      + a0 / L;
    out[oidx + 64] = hidden[oidx + 64] + a1 / L;
}

extern "C" void kernel_launch(void* const* d_in, const int* in_sizes, int n_in,
                              void* d_out, int out_size, void* d_ws, size_t ws_size,
                              hipStream_t stream) {
    const float* hidden = (const float*)d_in[0];
    const float* kcache = (const float*)d_in[1];
    const float* vcache = (const float*)d_in[2];
    const int*   btab   = (const int*)d_in[3];
    const int*   clen   = (const int*)d_in[4];
    const float* w      = (const float*)d_in[5];
    float*       out    = (float*)d_out;
    float*       qn     = (float*)d_ws;   // B*HIDDEN floats = 256 KB scratch

    rmsnorm_q_kernel<<<16, 256, 0, stream>>>(hidden, w, qn);
    paged_attn_kernel<<<16 * 32, NW * 32, 0, stream>>>(qn, kcache, vcache, btab, clen, hidden, out);
}